// GRUSetModel_87917980549562
// MI455X (gfx1250) — compile-verified
//
#include <hip/hip_runtime.h>
#include <math.h>

// ---------------- model dims ----------------
#define NPLY 1024
#define SEQ  1026          // env token + 1024 players + fund token
#define HDIM 128
#define NH   4
#define DH   32
#define INF_ 18
#define LN_EPS 1e-5f

typedef __attribute__((ext_vector_type(2))) float v2f;
typedef __attribute__((ext_vector_type(8))) float v8f;

__device__ __forceinline__ float sigmoidf_(float x) { return 1.0f / (1.0f + expf(-x)); }

// =====================================================================
// Generic fp32 WMMA GEMM with N-register-blocking + staged loads:
//   D[M,N] = act(alpha * X(MxK) * Op(W) + bias + res)
//   TRANS_B = true :  Op(W) = W^T, W is (N x K) row-major (ldw)   [X @ W^T]
//   TRANS_B = false:  Op(W) = W,   W is (K x N) row-major (ldw)   [X @ W]
//   NT      = number of 16-wide N tiles per wave (A fragment reused NT x)
//   KTAIL   = true iff K % 16 != 0 (guarded per-4 loop with exact A zero-fill)
// One wave32 per 16 x (16*NT) output tile; K consumed 4/instr via
// V_WMMA_F32_16X16X4_F32 (fp32 in / fp32 accumulate).
// KTAIL=false path: 16 K-steps per iteration, loads staged into registers
// first, then 4*NT back-to-back WMMAs -> load clause overlaps matrix pipe.
// =====================================================================
template <bool TRANS_B, int NT, bool KTAIL>
__global__ void __launch_bounds__(32)
wmma_gemm(const float* __restrict__ X, int ldx,
          const float* __restrict__ W, int ldw,
          float* __restrict__ D, int ldd,
          const float* __restrict__ bias,
          const float* __restrict__ res, int ldr,
          int M, int N, int K, float alpha, int do_relu)
{
    const int lane = threadIdx.x;         // 0..31, wave32
    const int half = lane >> 4;           // 0 or 1  (K-pair select)
    const int lid  = lane & 15;           // row-of-A / col-of-B
    const int m  = blockIdx.x * 16 + lid;
    const int mc = (m < M) ? m : (M - 1);

    int nc[NT];
#pragma unroll
    for (int t = 0; t < NT; ++t) {
        const int n = blockIdx.y * (16 * NT) + t * 16 + lid;
        nc[t] = (n < N) ? n : (N - 1);
    }

    const v8f vzero = {};
    v8f acc[NT];
#pragma unroll
    for (int t = 0; t < NT; ++t) acc[t] = vzero;

    const float* __restrict__ Xrow = X + (size_t)mc * ldx;
    __builtin_prefetch(Xrow, 0, 0);                          // global_prefetch_b8
    if (TRANS_B) __builtin_prefetch(W + (size_t)nc[0] * ldw, 0, 0);

    const int kb = half * 2;

    if (!KTAIL) {
        // ---- pipelined path: requires K % 16 == 0 ----
        for (int k0 = 0; k0 < K; k0 += 16) {
            v2f af[4];
            v2f bf[4][NT];
            // Phase 1: stage all fragments for 4 K-substeps (load clause)
#pragma unroll
            for (int u = 0; u < 4; ++u) {
                const int ka = k0 + 4 * u + kb;
                af[u].x = Xrow[ka];
                af[u].y = Xrow[ka + 1];
#pragma unroll
                for (int t = 0; t < NT; ++t) {
                    if (TRANS_B) {
                        bf[u][t].x = W[(size_t)nc[t] * ldw + ka];
                        bf[u][t].y = W[(size_t)nc[t] * ldw + ka + 1];
                    } else {
                        bf[u][t].x = W[(size_t)ka * ldw + nc[t]];
                        bf[u][t].y = W[(size_t)(ka + 1) * ldw + nc[t]];
                    }
                }
            }
            // Phase 2: drain with 4*NT back-to-back matrix ops
#pragma unroll
            for (int u = 0; u < 4; ++u) {
#pragma unroll
                for (int t = 0; t < NT; ++t) {
                    // (neg_a, A, neg_b, B, c_mod, C, reuse_a, reuse_b)
                    acc[t] = __builtin_amdgcn_wmma_f32_16x16x4_f32(
                        false, af[u], false, bf[u][t], (short)0, acc[t],
                        false, false);
                }
            }
        }
    } else {
        // ---- guarded path: arbitrary K ----
        for (int k0 = 0; k0 < K; k0 += 4) {
            const int ka0 = k0 + kb;
            const int ka1 = ka0 + 1;
            const int ca0 = (ka0 < K) ? ka0 : (K - 1);   // clamp: in-bounds
            const int ca1 = (ka1 < K) ? ka1 : (K - 1);

            float a0 = Xrow[ca0];
            float a1 = Xrow[ca1];
            if (ka0 >= K) a0 = 0.0f;   // zero A K-tail -> exact (kills k-term)
            if (ka1 >= K) a1 = 0.0f;
            v2f av; av.x = a0; av.y = a1;

#pragma unroll
            for (int t = 0; t < NT; ++t) {
                float b0, b1;
                if (TRANS_B) {
                    b0 = W[(size_t)nc[t] * ldw + ca0];
                    b1 = W[(size_t)nc[t] * ldw + ca1];
                } else {
                    b0 = W[(size_t)ca0 * ldw + nc[t]];
                    b1 = W[(size_t)ca1 * ldw + nc[t]];
                }
                v2f bv; bv.x = b0; bv.y = b1;
                acc[t] = __builtin_amdgcn_wmma_f32_16x16x4_f32(
                    false, av, false, bv, (short)0, acc[t], false, false);
            }
        }
    }

    // Epilogue: C/D layout -> VGPR v: row = v + 8*half, col = lid
#pragma unroll
    for (int t = 0; t < NT; ++t) {
        const int col = blockIdx.y * (16 * NT) + t * 16 + lid;
#pragma unroll
        for (int v = 0; v < 8; ++v) {
            const int row = blockIdx.x * 16 + v + half * 8;
            if (row < M && col < N) {
                float val = alpha * acc[t][v];
                if (bias) val += bias[col];
                if (res)  val += res[(size_t)row * ldr + col];
                if (do_relu) val = fmaxf(val, 0.0f);
                D[(size_t)row * ldd + col] = val;
            }
        }
    }
}

// =====================================================================
// Token embedding: h[s][c] = x[s] . fc_w[c] + fc_b[c]  (x built on the fly)
// =====================================================================
__global__ void build_h_kernel(const float* __restrict__ env,
                               const float* __restrict__ pf,
                               const float* __restrict__ mw,
                               const float* __restrict__ lc,
                               const float* __restrict__ wal,
                               const float* __restrict__ fc_w,
                               const float* __restrict__ fc_b,
                               float* __restrict__ h)
{
    int i = blockIdx.x * blockDim.x + threadIdx.x;
    if (i >= SEQ * HDIM) return;
    int s = i / HDIM, c = i % HDIM;
    const float* wrow = fc_w + c * INF_;
    float acc = fc_b[c];
    if (s == 0) {
#pragma unroll
        for (int k = 2; k < INF_; ++k) acc += env[k - 2] * wrow[k];
    } else if (s == SEQ - 1) {
        acc += pf[0] * wrow[0] + mw[0] * wrow[1];
    } else {
        acc += lc[s - 1] * wrow[0] + wal[s - 1] * wrow[1];
    }
    h[i] = acc;
}

// =====================================================================
// Row softmax (in place), one 256-thread block per row
// =====================================================================
__global__ void __launch_bounds__(256)
softmax_rows(float* __restrict__ P, int cols)
{
    __shared__ float red[256];
    float* row = P + (size_t)blockIdx.x * cols;
    const int t = threadIdx.x;

    float mx = -3.4e38f;
    for (int j = t; j < cols; j += 256) mx = fmaxf(mx, row[j]);
    red[t] = mx; __syncthreads();
    for (int s = 128; s > 0; s >>= 1) { if (t < s) red[t] = fmaxf(red[t], red[t + s]); __syncthreads(); }
    mx = red[0]; __syncthreads();

    float sum = 0.0f;
    for (int j = t; j < cols; j += 256) { float e = expf(row[j] - mx); row[j] = e; sum += e; }
    red[t] = sum; __syncthreads();
    for (int s = 128; s > 0; s >>= 1) { if (t < s) red[t] += red[t + s]; __syncthreads(); }
    const float inv = 1.0f / red[0];
    for (int j = t; j < cols; j += 256) row[j] *= inv;
}

// =====================================================================
// out = LayerNorm(a + b) * g + be ; one 128-thread block per row
// =====================================================================
__global__ void __launch_bounds__(128)
addln_kernel(const float* __restrict__ a, const float* __restrict__ b,
             const float* __restrict__ g, const float* __restrict__ be,
             float* __restrict__ out)
{
    __shared__ float red[128];
    const int r = blockIdx.x, c = threadIdx.x;
    const float x = a[(size_t)r * HDIM + c] + b[(size_t)r * HDIM + c];
    red[c] = x; __syncthreads();
    for (int s = 64; s > 0; s >>= 1) { if (c < s) red[c] += red[c + s]; __syncthreads(); }
    const float mu = red[0] * (1.0f / HDIM); __syncthreads();
    const float d = x - mu;
    red[c] = d * d; __syncthreads();
    for (int s = 64; s > 0; s >>= 1) { if (c < s) red[c] += red[c + s]; __syncthreads(); }
    const float var = red[0] * (1.0f / HDIM);
    out[(size_t)r * HDIM + c] = d * rsqrtf(var + LN_EPS) * g[c] + be[c];
}

// =====================================================================
// prev[i][:] = memory[idx[i]][:]
// =====================================================================
__global__ void gather_kernel(const float* __restrict__ mem,
                              const int* __restrict__ idx,
                              float* __restrict__ out)
{
    int i = blockIdx.x * blockDim.x + threadIdx.x;
    if (i >= NPLY * HDIM) return;
    int r = i / HDIM, c = i % HDIM;
    out[i] = mem[(size_t)idx[r] * HDIM + c];
}

// =====================================================================
// GRU gates (elementwise)
// =====================================================================
__global__ void gru_kernel(const float* __restrict__ gx, const float* __restrict__ gh,
                           const float* __restrict__ prev, float* __restrict__ memout)
{
    int i = blockIdx.x * blockDim.x + threadIdx.x;
    if (i >= NPLY * HDIM) return;
    int r = i / HDIM, c = i % HDIM;
    size_t b = (size_t)r * 3 * HDIM;
    float rg = sigmoidf_(gx[b + c] + gh[b + c]);
    float zg = sigmoidf_(gx[b + HDIM + c] + gh[b + HDIM + c]);
    float ng = tanhf(gx[b + 2 * HDIM + c] + rg * gh[b + 2 * HDIM + c]);
    memout[i] = (1.0f - zg) * ng + zg * prev[i];
}

// =====================================================================
// c_hat[r] = mem[r] . hc_w + hc_b
// =====================================================================
__global__ void __launch_bounds__(128)
chat_kernel(const float* __restrict__ mem, const float* __restrict__ hw,
            const float* __restrict__ hb, float* __restrict__ out)
{
    __shared__ float red[128];
    const int r = blockIdx.x, c = threadIdx.x;
    red[c] = mem[(size_t)r * HDIM + c] * hw[c];
    __syncthreads();
    for (int s = 64; s > 0; s >>= 1) { if (c < s) red[c] += red[c + s]; __syncthreads(); }
    if (c == 0) out[r] = red[0] + hb[0];
}

// =====================================================================
// Pairwise edge head: for each (i,j):
//   t = relu(hi[i]+hj[j]+e1_b);  p = sigmoid(t.w0+b0);  u = relu(t.w1+b1)
// 32x32 pair tile per block, hi/hj tiles staged in LDS (padded rows so
// lane-varying row index hits distinct banks).
// =====================================================================
__global__ void __launch_bounds__(256)
pair_kernel(const float* __restrict__ hi, const float* __restrict__ hj,
            const float* __restrict__ e1b,
            const float* __restrict__ e2w, const float* __restrict__ e2b,
            float* __restrict__ p_hat, float* __restrict__ u_hat)
{
    __shared__ float his[32][HDIM + 1];
    __shared__ float hjs[32][HDIM + 1];
    __shared__ float w0[HDIM], w1[HDIM], bs[HDIM];
    const int t = threadIdx.x;
    const int bi = blockIdx.x, bj = blockIdx.y;

    for (int e = t; e < 32 * HDIM; e += 256) {
        int rr = e / HDIM, cc = e % HDIM;
        his[rr][cc] = hi[(size_t)(bi * 32 + rr) * HDIM + cc];
        hjs[rr][cc] = hj[(size_t)(bj * 32 + rr) * HDIM + cc];
    }
    if (t < HDIM) { w0[t] = e2w[t]; w1[t] = e2w[HDIM + t]; bs[t] = e1b[t]; }
    __syncthreads();

    const float b0 = e2b[0], b1 = e2b[1];
    for (int p = t; p < 1024; p += 256) {
        const int li = p >> 5, lj = p & 31;
        float a0 = 0.0f, a1 = 0.0f;
#pragma unroll 4
        for (int c = 0; c < HDIM; ++c) {
            float v = fmaxf(his[li][c] + hjs[lj][c] + bs[c], 0.0f);
            a0 += v * w0[c];
            a1 += v * w1[c];
        }
        const int gi = bi * 32 + li, gj = bj * 32 + lj;
        const size_t o = (size_t)gi * NPLY + gj;
        p_hat[o] = sigmoidf_(a0 + b0);
        u_hat[o] = fmaxf(a1 + b1, 0.0f);
    }
}

// =====================================================================
// Host-side orchestration
// =====================================================================
extern "C" void kernel_launch(void* const* d_in, const int* in_sizes, int n_in,
                              void* d_out, int out_size, void* d_ws, size_t ws_size,
                              hipStream_t stream)
{
    (void)in_sizes; (void)n_in; (void)out_size; (void)ws_size;

    const float* env     = (const float*)d_in[0];
    const float* pf      = (const float*)d_in[1];
    const float* mw      = (const float*)d_in[2];
    const float* lc      = (const float*)d_in[3];
    const float* wal     = (const float*)d_in[4];
    const float* memory  = (const float*)d_in[5];
    const int*   pidx    = (const int*)  d_in[6];
    const float* fc_w    = (const float*)d_in[7];
    const float* fc_b    = (const float*)d_in[8];
    const float* in_w    = (const float*)d_in[9];
    const float* in_b    = (const float*)d_in[10];
    const float* out_w   = (const float*)d_in[11];
    const float* out_b   = (const float*)d_in[12];
    const float* ln_g    = (const float*)d_in[13];
    const float* ln_b    = (const float*)d_in[14];
    const float* ff1_w   = (const float*)d_in[15];
    const float* ff1_b   = (const float*)d_in[16];
    const float* ff2_w   = (const float*)d_in[17];
    const float* ff2_b   = (const float*)d_in[18];
    const float* gru_wi  = (const float*)d_in[19];
    const float* gru_wh  = (const float*)d_in[20];
    const float* gru_bi  = (const float*)d_in[21];
    const float* gru_bh  = (const float*)d_in[22];
    const float* hc_w    = (const float*)d_in[23];
    const float* hc_b    = (const float*)d_in[24];
    const float* e1_w    = (const float*)d_in[25];
    const float* e1_b    = (const float*)d_in[26];
    const float* e2_w    = (const float*)d_in[27];
    const float* e2_b    = (const float*)d_in[28];

    // ---- workspace layout (fp32) ----
    float* ws      = (float*)d_ws;
    float* hbuf    = ws;                                  // SEQ*H
    float* qkv     = hbuf   + (size_t)SEQ * HDIM;         // SEQ*3H
    float* scores  = qkv    + (size_t)SEQ * 3 * HDIM;     // NH*SEQ*SEQ
    float* attncat = scores + (size_t)NH * SEQ * SEQ;     // SEQ*H
    float* attnp   = attncat+ (size_t)SEQ * HDIM;         // SEQ*H
    float* h2      = attnp  + (size_t)SEQ * HDIM;         // SEQ*H
    float* ff1b    = h2     + (size_t)SEQ * HDIM;         // SEQ*H
    float* h3      = ff1b   + (size_t)SEQ * HDIM;         // SEQ*H
    float* prevb   = h3     + (size_t)SEQ * HDIM;         // N*H
    float* gx      = prevb  + (size_t)NPLY * HDIM;        // N*3H
    float* gh      = gx     + (size_t)NPLY * 3 * HDIM;    // N*3H
    float* membuf  = gh     + (size_t)NPLY * 3 * HDIM;    // N*H
    float* hib     = membuf + (size_t)NPLY * HDIM;        // N*H
    float* hjb     = hib    + (size_t)NPLY * HDIM;        // N*H

    float* c_hat = (float*)d_out;                          // N
    float* p_hat = c_hat + NPLY;                           // N*N
    float* u_hat = p_hat + (size_t)NPLY * NPLY;            // N*N

    const dim3 wave(32);
    const int TM  = (SEQ + 15) / 16;   // 65 row tiles for SEQ
    const int TMp = NPLY / 16;         // 64 row tiles for N players

    // 1) token embed
    build_h_kernel<<<(SEQ * HDIM + 255) / 256, 256, 0, stream>>>(
        env, pf, mw, lc, wal, fc_w, fc_b, hbuf);

    // 2) qkv = h @ in_w^T + in_b      (M=1026, N=384, K=128)
    wmma_gemm<true, 4, false><<<dim3(TM, 6), wave, 0, stream>>>(
        hbuf, HDIM, in_w, HDIM, qkv, 3 * HDIM, in_b, nullptr, 0,
        SEQ, 3 * HDIM, HDIM, 1.0f, 0);

    // 3) per-head scores = (q @ k^T)/sqrt(DH)   (M=N=1026, K=32)
    const float scal = 0.17677669529663687f;  // 1/sqrt(32)
    for (int hd = 0; hd < NH; ++hd) {
        wmma_gemm<true, 4, false><<<dim3(TM, (SEQ + 63) / 64), wave, 0, stream>>>(
            qkv + hd * DH, 3 * HDIM,            // q rows
            qkv + HDIM + hd * DH, 3 * HDIM,     // k rows (as W, transposed)
            scores + (size_t)hd * SEQ * SEQ, SEQ,
            nullptr, nullptr, 0, SEQ, SEQ, DH, scal, 0);
    }

    // 4) softmax over keys
    softmax_rows<<<NH * SEQ, 256, 0, stream>>>(scores, SEQ);

    // 5) per-head attn = P @ v   (M=1026, N=32, K=1026 -> K tail!)
    for (int hd = 0; hd < NH; ++hd) {
        wmma_gemm<false, 2, true><<<dim3(TM, 1), wave, 0, stream>>>(
            scores + (size_t)hd * SEQ * SEQ, SEQ,
            qkv + 2 * HDIM + hd * DH, 3 * HDIM,
            attncat + hd * DH, HDIM,
            nullptr, nullptr, 0, SEQ, DH, SEQ, 1.0f, 0);
    }

    // 6) out-projection, then h2 = LN(h + attnp)
    wmma_gemm<true, 4, false><<<dim3(TM, 2), wave, 0, stream>>>(
        attncat, HDIM, out_w, HDIM, attnp, HDIM, out_b, nullptr, 0,
        SEQ, HDIM, HDIM, 1.0f, 0);
    addln_kernel<<<SEQ, 128, 0, stream>>>(hbuf, attnp, ln_g, ln_b, h2);

    // 7) FF: h3 = relu(h2 @ ff1^T + b1) @ ff2^T + b2 + h2
    wmma_gemm<true, 4, false><<<dim3(TM, 2), wave, 0, stream>>>(
        h2, HDIM, ff1_w, HDIM, ff1b, HDIM, ff1_b, nullptr, 0,
        SEQ, HDIM, HDIM, 1.0f, 1);
    wmma_gemm<true, 4, false><<<dim3(TM, 2), wave, 0, stream>>>(
        ff1b, HDIM, ff2_w, HDIM, h3, HDIM, ff2_b, h2, HDIM,
        SEQ, HDIM, HDIM, 1.0f, 0);

    // 8) GRU: emb = h3 rows 1..N
    gather_kernel<<<(NPLY * HDIM + 255) / 256, 256, 0, stream>>>(memory, pidx, prevb);
    wmma_gemm<true, 4, false><<<dim3(TMp, 6), wave, 0, stream>>>(
        h3 + HDIM, HDIM, gru_wi, HDIM, gx, 3 * HDIM, gru_bi, nullptr, 0,
        NPLY, 3 * HDIM, HDIM, 1.0f, 0);
    wmma_gemm<true, 4, false><<<dim3(TMp, 6), wave, 0, stream>>>(
        prevb, HDIM, gru_wh, HDIM, gh, 3 * HDIM, gru_bh, nullptr, 0,
        NPLY, 3 * HDIM, HDIM, 1.0f, 0);
    gru_kernel<<<(NPLY * HDIM + 255) / 256, 256, 0, stream>>>(gx, gh, prevb, membuf);

    // 9) c_hat
    chat_kernel<<<NPLY, 128, 0, stream>>>(membuf, hc_w, hc_b, c_hat);

    // 10) hi = mem @ wl^T, hj = mem @ wr^T  (e1_w is (H x 2H) row-major)
    wmma_gemm<true, 4, false><<<dim3(TMp, 2), wave, 0, stream>>>(
        membuf, HDIM, e1_w, 2 * HDIM, hib, HDIM, nullptr, nullptr, 0,
        NPLY, HDIM, HDIM, 1.0f, 0);
    wmma_gemm<true, 4, false><<<dim3(TMp, 2), wave, 0, stream>>>(
        membuf, HDIM, e1_w + HDIM, 2 * HDIM, hjb, HDIM, nullptr, nullptr, 0,
        NPLY, HDIM, HDIM, 1.0f, 0);

    // 11) pairwise edge outputs
    pair_kernel<<<dim3(NPLY / 32, NPLY / 32), 256, 0, stream>>>(
        hib, hjb, e1_b, e2_w, e2_b, p_hat, u_hat);
}